// GNCA_42245298323835
// MI455X (gfx1250) — compile-verified
//
#include <hip/hip_runtime.h>
#include <hip/hip_bf16.h>

#define CCH 5
#define RADIUS_F 0.05f
#define ACC_SCALE_F 0.4f
#define MAX_VEL_F 0.1f

typedef __attribute__((ext_vector_type(2))) float v2f;
typedef __attribute__((ext_vector_type(8))) float v8f;

// Only check for the builtin in the DEVICE pass; the host x86 pass never has it.
#if defined(__HIP_DEVICE_COMPILE__) && !__has_builtin(__builtin_amdgcn_wmma_f32_16x16x4_f32)
#error "DEVICE pass lacks __builtin_amdgcn_wmma_f32_16x16x4_f32 (f32 16x16x4 WMMA)"
#endif

#if defined(__HIP_DEVICE_COMPILE__)
#define WMMA_F32_4(acc, a, b) \
  __builtin_amdgcn_wmma_f32_16x16x4_f32(false, (a), false, (b), (short)0, (acc), false, false)
#else
#define WMMA_F32_4(acc, a, b) (acc)
#endif

// Combined message matrix M[k][o] (k=0..15 rows, o=0..15 cols; only k<15,o<5 nonzero):
//   k = h*5+i  (h<2)  -> w_e2[h*25 + i*5 + o]
//   k = 10+i           -> b_e2[i*5 + o]
__device__ __forceinline__ float m_val(int k, int o,
                                       const float* __restrict__ w_e2,
                                       const float* __restrict__ b_e2) {
  if (o >= CCH || k >= 15) return 0.0f;
  if (k < 10) return w_e2[(k / CCH) * (CCH * CCH) + (k % CCH) * CCH + o];
  return b_e2[(k - 10) * CCH + o];
}

// ---------------------------------------------------------------------------
// Init: zero edge-aggregation buffer, packed degree counters, scalar outputs.
// ---------------------------------------------------------------------------
__global__ void gnca_init_kernel(float* __restrict__ agg,
                                 unsigned int* __restrict__ deg,
                                 float* __restrict__ out, int nN) {
  long long i = (long long)blockIdx.x * blockDim.x + threadIdx.x;
  long long tot = (long long)nN * CCH;
  if (i < tot) agg[i] = 0.0f;
  if (i < nN) deg[i] = 0u;
  if (i < CCH) out[tot + i] = 0.0f;  // vel_bonus[2], border, food, dead
}

// ---------------------------------------------------------------------------
// Edge pass: per 16-edge tile, msg = f[16x15] @ M[15x5] via 4x V_WMMA_F32_16X16X4_F32
// (exact f32), then scatter-atomic into agg[dst]. One packed u32 atomic per edge
// carries both degree counters (near in [15:0], cell-edge in [31:16]).
// ---------------------------------------------------------------------------
__global__ void __launch_bounds__(256) gnca_edge_kernel(
    const float* __restrict__ x, const int* __restrict__ ei,
    const float* __restrict__ ea,
    const float* __restrict__ w_e1, const float* __restrict__ b_e1,
    const float* __restrict__ w_e2, const float* __restrict__ b_e2,
    float* __restrict__ agg, unsigned int* __restrict__ deg,
    int nE, int nWaves) {
  const int lane = threadIdx.x & 31;
  const int wave = blockIdx.x * (blockDim.x >> 5) + (threadIdx.x >> 5);
  const int n = lane & 15;        // edge-in-tile (A build) / output channel (scatter)
  const bool hi = lane >= 16;

  // Edge-MLP layer-1 weights (wave-uniform scalar loads). w_e1 is [2,2] row-major.
  const float w00 = w_e1[0], w01 = w_e1[1], w10 = w_e1[2], w11 = w_e1[3];
  const float be0 = b_e1[0], be1 = b_e1[1];

  // Loop-invariant B operand: lanes 0-15 -> col n with K rows {4j,4j+1};
  // lanes 16-31 -> col n with K rows {4j+2,4j+3} (mirrors the ISA A layout).
  const int kh = hi ? 2 : 0;
  v2f B0, B1, B2, B3;
  B0.x = m_val(0 + kh, n, w_e2, b_e2);  B0.y = m_val(1 + kh, n, w_e2, b_e2);
  B1.x = m_val(4 + kh, n, w_e2, b_e2);  B1.y = m_val(5 + kh, n, w_e2, b_e2);
  B2.x = m_val(8 + kh, n, w_e2, b_e2);  B2.y = m_val(9 + kh, n, w_e2, b_e2);
  B3.x = m_val(12 + kh, n, w_e2, b_e2); B3.y = m_val(13 + kh, n, w_e2, b_e2);

  const long long stride = (long long)nWaves * 16;
  for (long long base = (long long)wave * 16; base < nE; base += stride) {
    // Prefetch next tile of edge_attr (global_prefetch_b8).
    {
      const long long pe = base + stride + n;
      if (pe < nE) __builtin_prefetch(&ea[2 * pe], 0, 0);
    }

    const int e = (int)base + n;            // both half-waves mirror the same edge
    const bool valid = e < nE;
    float d0 = 0.0f, d1 = 0.0f;
    float xs0 = 0.0f, xs1 = 0.0f, xs2 = 0.0f, xs3 = 0.0f, xs4 = 0.0f;
    int src = 0;
    if (valid) {
      const float2 eav = ((const float2*)ea)[e];
      d0 = eav.x;
      d1 = eav.y;
      src = ei[e];  // edge_index row 0 (src)
      const float* xp = x + (long long)src * CCH;  // x fits in L2 (10 MB)
      xs0 = xp[0]; xs1 = xp[1]; xs2 = xp[2]; xs3 = xp[3]; xs4 = xp[4];
    }

    // Hoisted dst-index loads: all 16 indices sit in one 64B line; clamp
    // instead of branching so the 8 loads issue as one clause, one wait.
    int dsts[8];
#pragma unroll
    for (int r = 0; r < 8; ++r) {
      int er = (int)base + r + (hi ? 8 : 0);
      er = er < nE ? er : nE - 1;
      dsts[r] = ei[(long long)nE + er];  // edge_index row 1 (dst)
    }

    const float h0 = fmaxf(fmaf(d1, w10, fmaf(d0, w00, be0)), 0.0f);
    const float h1 = fmaxf(fmaf(d1, w11, fmaf(d0, w01, be1)), 0.0f);

    // Feature vector f[k] of this lane's edge, as NAMED scalars (no array ->
    // no dynamic-index select trees; each A element is exactly one v_cndmask).
    // k: 0..4 = h0*xs, 5..9 = h1*xs, 10..14 = xs, 15 = 0 (K padding).
    const float f0 = h0 * xs0, f1 = h0 * xs1, f2 = h0 * xs2, f3 = h0 * xs3, f4 = h0 * xs4;
    const float f5 = h1 * xs0, f6 = h1 * xs1, f7 = h1 * xs2, f8 = h1 * xs3, f9 = h1 * xs4;
    // f10..f14 == xs0..xs4

    v2f a0, a1, a2, a3;
    a0.x = hi ? f2  : f0;   a0.y = hi ? f3   : f1;   // K {0,1} | {2,3}
    a1.x = hi ? f6  : f4;   a1.y = hi ? f7   : f5;   // K {4,5} | {6,7}
    a2.x = hi ? xs0 : f8;   a2.y = hi ? xs1  : f9;   // K {8,9} | {10,11}
    a3.x = hi ? xs4 : xs2;  a3.y = hi ? 0.0f : xs3;  // K {12,13} | {14,15}

    // D = sum_j A_j * B_j  (EXEC is all-ones here: no divergence at this point)
    v8f acc = {0.0f, 0.0f, 0.0f, 0.0f, 0.0f, 0.0f, 0.0f, 0.0f};
    acc = WMMA_F32_4(acc, a0, B0);
    acc = WMMA_F32_4(acc, a1, B1);
    acc = WMMA_F32_4(acc, a2, B2);
    acc = WMMA_F32_4(acc, a3, B3);

    // Packed degree counters at src (low half only -> once per edge).
    if (valid && !hi) {
      const unsigned int pack =
          (d0 < RADIUS_F ? 1u : 0u) | (d1 == 1.0f ? 0x10000u : 0u);
      atomicAdd(&deg[src], pack);
    }

    // Scatter: VGPR r of D holds edge (base + r + 8*hi), channel = n.
#pragma unroll
    for (int r = 0; r < 8; ++r) {
      const int er = (int)base + r + (hi ? 8 : 0);
      if (n < CCH && er < nE) {
        atomicAdd(&agg[(long long)dsts[r] * CCH + n], acc[r]);
      }
    }
  }
}

// ---------------------------------------------------------------------------
// Node pass: h = agg + x@root + b ; mlp2 ; acc/vel/pos ; masks ; reductions.
// ---------------------------------------------------------------------------
__global__ void __launch_bounds__(256) gnca_node_kernel(
    const float* __restrict__ x, const float* __restrict__ agg,
    const unsigned int* __restrict__ deg,
    const float* __restrict__ root, const float* __restrict__ conv_b,
    const float* __restrict__ w_h1, const float* __restrict__ b_h1,
    const float* __restrict__ w_h2, const float* __restrict__ b_h2,
    float* __restrict__ out, int nN) {
  __shared__ float sred[5];  // |v0|, |v1|, border, food, dead
  if (threadIdx.x < 5) sred[threadIdx.x] = 0.0f;
  __syncthreads();

  const int i = blockIdx.x * blockDim.x + threadIdx.x;
  float v0a = 0.0f, v1a = 0.0f, bc = 0.0f, fr = 0.0f, dc = 0.0f;
  if (i < nN) {
    float xv[CCH], h[CCH], t[CCH];
#pragma unroll
    for (int c = 0; c < CCH; ++c) xv[c] = x[(long long)i * CCH + c];
#pragma unroll
    for (int o = 0; o < CCH; ++o) {
      float s = agg[(long long)i * CCH + o] + conv_b[o];
#pragma unroll
      for (int k = 0; k < CCH; ++k) s = fmaf(xv[k], root[k * CCH + o], s);
      h[o] = s;
    }
#pragma unroll
    for (int o = 0; o < CCH; ++o) {
      float s = b_h1[o];
#pragma unroll
      for (int k = 0; k < CCH; ++k) s = fmaf(h[k], w_h1[k * CCH + o], s);
      t[o] = fmaxf(s, 0.0f);
    }
    float a0 = b_h2[0], a1 = b_h2[1];
#pragma unroll
    for (int o = 0; o < CCH; ++o) {
      a0 = fmaf(t[o], w_h2[o * 2 + 0], a0);
      a1 = fmaf(t[o], w_h2[o * 2 + 1], a1);
    }
    const float cellf = (xv[4] == 1.0f) ? 1.0f : 0.0f;
    a0 *= ACC_SCALE_F * cellf;
    a1 *= ACC_SCALE_F * cellf;
    const float vel0 = fminf(fmaxf(xv[2] + a0, -MAX_VEL_F), MAX_VEL_F);
    const float vel1 = fminf(fmaxf(xv[3] + a1, -MAX_VEL_F), MAX_VEL_F);
    const float p0 = xv[0] + vel0, p1 = xv[1] + vel1;
    const float ax = fabsf(p0), ay = fabsf(p1);
    if (ax > 1.0f) bc += logf(ax + 1e-6f);
    if (ay > 1.0f) bc += logf(ay + 1e-6f);

    const unsigned int d = deg[i];
    const int nearc = (int)(d & 0xffffu);
    const int cellc = (int)(d >> 16);
    const bool dead = (xv[4] == 1.0f) && (cellc < 1);
    const bool consume = (xv[4] == 0.0f) && (nearc >= 3);

    float* nx = out + (long long)i * CCH;
    nx[0] = p0; nx[1] = p1; nx[2] = vel0; nx[3] = vel1; nx[4] = xv[4];
    out[(long long)nN * CCH + CCH + i] = (dead || consume) ? 1.0f : 0.0f;

    v0a = fabsf(vel0);
    v1a = fabsf(vel1);
    fr = consume ? 1.0f : 0.0f;
    dc = dead ? 1.0f : 0.0f;
  }
  atomicAdd(&sred[0], v0a);             // ds_add_f32 (LDS atomics)
  atomicAdd(&sred[1], v1a);
  if (bc != 0.0f) atomicAdd(&sred[2], bc);
  if (fr != 0.0f) atomicAdd(&sred[3], fr);
  if (dc != 0.0f) atomicAdd(&sred[4], dc);
  __syncthreads();
  if (threadIdx.x == 0) {
    const long long s = (long long)nN * CCH;
    atomicAdd(&out[s + 0], sred[0] / (float)nN);  // vel_bonus mean
    atomicAdd(&out[s + 1], sred[1] / (float)nN);
    if (sred[2] != 0.0f) atomicAdd(&out[s + 2], sred[2]);  // border_cost
    if (sred[3] != 0.0f) atomicAdd(&out[s + 3], sred[3]);  // food_reward
    if (sred[4] != 0.0f) atomicAdd(&out[s + 4], sred[4]);  // dead_cost
  }
}

// ---------------------------------------------------------------------------
extern "C" void kernel_launch(void* const* d_in, const int* in_sizes, int n_in,
                              void* d_out, int out_size, void* d_ws, size_t ws_size,
                              hipStream_t stream) {
  const float* x      = (const float*)d_in[0];
  const int*   ei     = (const int*)d_in[1];
  const float* ea     = (const float*)d_in[2];
  const float* w_e1   = (const float*)d_in[3];
  const float* b_e1   = (const float*)d_in[4];
  const float* w_e2   = (const float*)d_in[5];
  const float* b_e2   = (const float*)d_in[6];
  const float* root   = (const float*)d_in[7];
  const float* conv_b = (const float*)d_in[8];
  const float* w_h1   = (const float*)d_in[9];
  const float* b_h1   = (const float*)d_in[10];
  const float* w_h2   = (const float*)d_in[11];
  const float* b_h2   = (const float*)d_in[12];
  float* out = (float*)d_out;

  const int nN = in_sizes[0] / CCH;
  const int nE = in_sizes[1] / 2;

  float* agg = (float*)d_ws;                                   // [nN * 5] f32
  unsigned int* deg =
      (unsigned int*)((char*)d_ws + (size_t)nN * CCH * sizeof(float));  // [nN] u32

  {
    const long long tot = (long long)nN * CCH;
    const int blocks = (int)((tot + 255) / 256);
    gnca_init_kernel<<<blocks, 256, 0, stream>>>(agg, deg, out, nN);
  }
  {
    const int blockSz = 256;                 // 8 wave32 per block
    const int blocks = 2048;
    const int nWaves = blocks * (blockSz / 32);
    gnca_edge_kernel<<<blocks, blockSz, 0, stream>>>(
        x, ei, ea, w_e1, b_e1, w_e2, b_e2, agg, deg, nE, nWaves);
  }
  {
    const int blocks = (nN + 255) / 256;
    gnca_node_kernel<<<blocks, 256, 0, stream>>>(
        x, agg, deg, root, conv_b, w_h1, b_h1, w_h2, b_h2, out, nN);
  }
}